// DeformableAttentionSampler_27358941675861
// MI455X (gfx1250) — compile-verified
//
#include <hip/hip_runtime.h>
#include <hip/hip_bf16.h>
#include <math.h>

// ---- problem constants (from reference) ----
#define EMBED   256
#define NHEAD   8
#define NPNT    32
#define STRIDEV 4
#define FFND    1024
#define NUMIMG  4
#define HH      160
#define WW      160
#define NOBJ    2048
#define HD      32          // EMBED / NHEAD
#define LN_EPS  1e-5f

typedef float v2f __attribute__((ext_vector_type(2)));
typedef float v8f __attribute__((ext_vector_type(8)));

__device__ __forceinline__ int clampi(int v, int lo, int hi) {
    return v < lo ? lo : (v > hi ? hi : v);
}

// =====================================================================
// f32 WMMA GEMM: C(MxN) = A(MxK) @ B(KxN) + bias[N] (+ residual) (+ relu)
// K and N are compile-time so all address math is constant-offset (no
// runtime loop versioning). One wave32 computes a 16 x (16*NT) tile with
// V_WMMA_F32_16X16X4_F32 (K-step 4); the A fragment is reused across the
// NT n-subtiles, so each k-step costs 2 A-loads + 2*NT B-loads for NT
// WMMAs (vs 4 loads per WMMA with a single tile).
// Fragment layouts (ISA 7.12.2):
//   A 16x4 : lanes 0-15 -> M=lane,    vgpr0=K+0, vgpr1=K+1
//            lanes 16-31-> M=lane-16, vgpr0=K+2, vgpr1=K+3
//   B 4x16 : mirrored (row K striped across 16 lanes per half)
//   C/D    : vgpr j -> row m0 + j + 8*(lane>=16), col n0 + (lane&15)
// =====================================================================
template<int K, int N, int NT, int RELU, int RESID>
__global__ __launch_bounds__(128)
void gemm_wmma_f32(const float* __restrict__ A, const float* __restrict__ B,
                   const float* __restrict__ bias, const float* __restrict__ resid,
                   float* __restrict__ C)
{
    constexpr int groupsN = (N / 16) / NT;   // wave-tile columns
    const int lane = threadIdx.x & 31;
    const int wave = threadIdx.x >> 5;
    const int t    = blockIdx.x * 4 + wave;  // grid sized exactly
    const int m0   = (t / groupsN) * 16;
    const int n0   = (t % groupsN) * (16 * NT);
    const int half = lane >> 4;
    const int lm   = lane & 15;

    const float* __restrict__ arow = A + (size_t)(m0 + lm) * K + 2 * half;
    const float* __restrict__ brow = B + (size_t)(2 * half) * N + n0 + lm;

    v8f acc[NT];
#pragma unroll
    for (int j = 0; j < NT; ++j) acc[j] = (v8f){};

    for (int k = 0; k < K; k += 4) {
        v2f a;
        a.x = arow[k + 0];
        a.y = arow[k + 1];
        const float* __restrict__ bk = brow + (size_t)k * N;
#pragma unroll
        for (int j = 0; j < NT; ++j) {
            v2f b;
            b.x = bk[j * 16];
            b.y = bk[j * 16 + N];
            acc[j] = __builtin_amdgcn_wmma_f32_16x16x4_f32(
                         false, a, false, b, (short)0, acc[j], false, false);
        }
    }

#pragma unroll
    for (int j = 0; j < NT; ++j) {
#pragma unroll
        for (int r = 0; r < 8; ++r) {
            const int row = m0 + r + 8 * half;
            const int col = n0 + j * 16 + lm;
            float v = acc[j][r] + bias[col];
            if (RESID) v += resid[(size_t)row * N + col];
            if (RELU)  v = v > 0.0f ? v : 0.0f;
            C[(size_t)row * N + col] = v;
        }
    }
}

// =====================================================================
// Deformable sampling + 32-point softmax attention.
// One wave per (object, head); lane = sample point p. v_s staged in LDS
// (pad 33 -> conflict-free column reads) for the p<->c transpose used by
// both the attention output and the v_samples layout.
// =====================================================================
__global__ __launch_bounds__(256)
void sampler_kernel(const float* __restrict__ query,     // (N, NH, 1, HD)
                    const float* __restrict__ key_feat,  // (4, 256, H, W)
                    const float* __restrict__ value,     // (4, 256, H, W)
                    const float* __restrict__ x2d,       // (4, 2, H, W)
                    const float* __restrict__ maskf,     // (4, 1, H, W)
                    const float* __restrict__ obj_xy,    // (N, 2)
                    const float* __restrict__ strides,   // (N,)
                    const int*   __restrict__ img_ind,   // (N,)
                    const float* __restrict__ offs,      // (N, NH, NP, 2), bias included
                    float* __restrict__ att_out,         // (N, EMBED)
                    float* __restrict__ v_samples,       // (N, NH, HD, NP)
                    float* __restrict__ a_samples,       // (N, NH, 1, NP)
                    float* __restrict__ mask_samples,    // (N, NH, 1, NP)
                    float* __restrict__ x2d_samples)     // (N, NH, 2, NP)
{
    __shared__ float lds_v[8][NPNT * 33];
    __shared__ float lds_a[8][NPNT];

    const int wave = threadIdx.x >> 5;
    const int lane = threadIdx.x & 31;   // point index p
    const int wg   = blockIdx.x * 8 + wave;
    const int n    = wg >> 3;
    const int h    = wg & 7;
    const int p    = lane;

    // ---- sample location ----
    const size_t obase = (((size_t)n * NHEAD + h) * NPNT + p) * 2;
    const float st = strides[n];
    const float lx = obj_xy[n * 2 + 0] + offs[obase + 0] * st;
    const float ly = obj_xy[n * 2 + 1] + offs[obase + 1] * st;
    const float gx = lx * (2.0f / (WW * STRIDEV)) - 1.0f;
    const float gy = ly * (2.0f / (HH * STRIDEV)) - 1.0f;
    const float ix = ((gx + 1.0f) * WW - 1.0f) * 0.5f;
    const float iy = ((gy + 1.0f) * HH - 1.0f) * 0.5f;
    const float x0f = floorf(ix), y0f = floorf(iy);
    const float wx = ix - x0f, wy = iy - y0f;
    const int x0 = (int)x0f, y0 = (int)y0f;
    const int x1 = x0 + 1,   y1 = y0 + 1;
    const int x0c = clampi(x0, 0, WW - 1), x1c = clampi(x1, 0, WW - 1);
    const int y0c = clampi(y0, 0, HH - 1), y1c = clampi(y1, 0, HH - 1);
    const float w00 = (1.0f - wx) * (1.0f - wy), w10 = wx * (1.0f - wy);
    const float w01 = (1.0f - wx) * wy,          w11 = wx * wy;
    const int o00 = y0c * WW + x0c, o10 = y0c * WW + x1c;
    const int o01 = y1c * WW + x0c, o11 = y1c * WW + x1c;

    const int img = img_ind[n];
    const size_t plane = (size_t)HH * WW;
    const float* __restrict__ kb = key_feat + ((size_t)img * EMBED + h * HD) * plane;
    const float* __restrict__ vb = value    + ((size_t)img * EMBED + h * HD) * plane;
    const float* __restrict__ xb = x2d + (size_t)img * 2 * plane;
    const float* __restrict__ mb = maskf + (size_t)img * plane;
    const float* __restrict__ qb = query + ((size_t)n * NHEAD + h) * HD;

    // ---- gather K (dot with q on the fly) and V (stash to LDS) ----
    float a = 0.0f;
    for (int c = 0; c < HD; ++c) {
        const float* kc = kb + (size_t)c * plane;
        const float ks = w00 * kc[o00] + w10 * kc[o10] + w01 * kc[o01] + w11 * kc[o11];
        a += qb[c] * ks;
        const float* vc = vb + (size_t)c * plane;
        const float vs = w00 * vc[o00] + w10 * vc[o10] + w01 * vc[o01] + w11 * vc[o11];
        lds_v[wave][p * 33 + c] = vs;
    }
    a *= 0.17677669529663687f;  // 1/sqrt(HD)

    // ---- x2d (border pad) and mask (zeros pad) samples ----
    const float xs0 = w00 * xb[o00] + w10 * xb[o10] + w01 * xb[o01] + w11 * xb[o11];
    const float* xb1 = xb + plane;
    const float xs1 = w00 * xb1[o00] + w10 * xb1[o10] + w01 * xb1[o01] + w11 * xb1[o11];
    const float mw00 = (x0 >= 0 && x0 < WW && y0 >= 0 && y0 < HH) ? w00 : 0.0f;
    const float mw10 = (x1 >= 0 && x1 < WW && y0 >= 0 && y0 < HH) ? w10 : 0.0f;
    const float mw01 = (x0 >= 0 && x0 < WW && y1 >= 0 && y1 < HH) ? w01 : 0.0f;
    const float mw11 = (x1 >= 0 && x1 < WW && y1 >= 0 && y1 < HH) ? w11 : 0.0f;
    const float ms = mw00 * mb[o00] + mw10 * mb[o10] + mw01 * mb[o01] + mw11 * mb[o11];

    // ---- softmax across the 32 points (one per lane), wave32 shuffles ----
    float amax = a;
#pragma unroll
    for (int m = 16; m >= 1; m >>= 1) amax = fmaxf(amax, __shfl_xor(amax, m, 32));
    const float e = __expf(a - amax);
    float esum = e;
#pragma unroll
    for (int m = 16; m >= 1; m >>= 1) esum += __shfl_xor(esum, m, 32);
    const float aw = e / esum * ms;
    lds_a[wave][p] = aw;
    __syncthreads();

    // ---- attention output: lane = channel c ----
    const int c = lane;
    float oc = 0.0f;
#pragma unroll 4
    for (int pp = 0; pp < NPNT; ++pp)
        oc += lds_v[wave][pp * 33 + c] * lds_a[wave][pp];
    att_out[(size_t)n * EMBED + h * HD + c] = oc;

    // ---- sampled-tensor outputs ----
    const size_t vbase = (((size_t)n * NHEAD + h) * HD + c) * NPNT;
#pragma unroll 4
    for (int pp = 0; pp < NPNT; ++pp)
        v_samples[vbase + pp] = lds_v[wave][pp * 33 + c];

    const size_t pbase = ((size_t)n * NHEAD + h) * NPNT;
    a_samples[pbase + p]    = a;     // pre-softmax, scaled
    mask_samples[pbase + p] = ms;
    const size_t x2b = ((size_t)n * NHEAD + h) * 2 * NPNT;
    x2d_samples[x2b + p]        = xs0;
    x2d_samples[x2b + NPNT + p] = xs1;
}

// =====================================================================
// LayerNorm over EMBED=256: one wave per row, 8 elems per lane.
// =====================================================================
__global__ __launch_bounds__(256)
void layernorm_kernel(const float* __restrict__ x, const float* __restrict__ g,
                      const float* __restrict__ b, float* __restrict__ y)
{
    const int wave = threadIdx.x >> 5;
    const int lane = threadIdx.x & 31;
    const int row  = blockIdx.x * 8 + wave;
    const float* xr = x + (size_t)row * EMBED;

    float v[8];
    float s = 0.0f;
#pragma unroll
    for (int j = 0; j < 8; ++j) { v[j] = xr[j * 32 + lane]; s += v[j]; }
#pragma unroll
    for (int m = 16; m >= 1; m >>= 1) s += __shfl_xor(s, m, 32);
    const float mean = s * (1.0f / EMBED);

    float vs = 0.0f;
#pragma unroll
    for (int j = 0; j < 8; ++j) { const float d = v[j] - mean; vs += d * d; }
#pragma unroll
    for (int m = 16; m >= 1; m >>= 1) vs += __shfl_xor(vs, m, 32);
    const float rstd = rsqrtf(vs * (1.0f / EMBED) + LN_EPS);

    float* yr = y + (size_t)row * EMBED;
#pragma unroll
    for (int j = 0; j < 8; ++j) {
        const int col = j * 32 + lane;
        yr[col] = (v[j] - mean) * rstd * g[col] + b[col];
    }
}

// =====================================================================
extern "C" void kernel_launch(void* const* d_in, const int* in_sizes, int n_in,
                              void* d_out, int out_size, void* d_ws, size_t ws_size,
                              hipStream_t stream)
{
    (void)in_sizes; (void)n_in; (void)out_size; (void)ws_size;

    const float* query   = (const float*)d_in[0];
    const float* obj_emb = (const float*)d_in[1];
    const float* key_feat= (const float*)d_in[2];
    const float* value   = (const float*)d_in[3];
    const float* x2d     = (const float*)d_in[4];
    const float* maskf   = (const float*)d_in[5];
    const float* obj_xy  = (const float*)d_in[6];
    const float* strides = (const float*)d_in[7];
    const int*   img_ind = (const int*)d_in[8];
    const float* w_off   = (const float*)d_in[9];
    const float* b_off   = (const float*)d_in[10];
    const float* w_out   = (const float*)d_in[11];
    const float* b_out   = (const float*)d_in[12];
    const float* ln1_g   = (const float*)d_in[13];
    const float* ln1_b   = (const float*)d_in[14];
    const float* w1      = (const float*)d_in[15];
    const float* b1      = (const float*)d_in[16];
    const float* w2      = (const float*)d_in[17];
    const float* b2      = (const float*)d_in[18];
    const float* ln2_g   = (const float*)d_in[19];
    const float* ln2_b   = (const float*)d_in[20];

    // workspace layout (floats); hbuf (8MB) reuses offs|att|y1 which are dead
    float* ws   = (float*)d_ws;
    float* offs = ws;                          // 2048*512  = 1048576
    float* att  = ws + 1048576;                // 2048*256  =  524288
    float* y1   = ws + 1048576 + 524288;       // 2048*256
    float* hbuf = ws;                          // 2048*1024 = 2097152 (reuse)
    float* x1   = ws + 2097152;                // 2048*256
    float* y2   = ws + 2097152 + 524288;       // 2048*256

    // output tuple, concatenated flat
    float* out_final     = (float*)d_out;                 // 2048*256
    float* v_samples     = out_final + 524288;            // 2048*8*32*32
    float* a_samples     = v_samples + 2097152;           // 2048*8*32
    float* mask_samples  = a_samples + 524288;            // 2048*8*32
    float* x2d_samples   = mask_samples + 524288;         // 2048*8*2*32

    // waves per GEMM = (M/16) * (N/64); 4 waves per block, grids exact
    // 1) offsets = obj_emb @ w_off + b_off          (2048x512, K=256)
    gemm_wmma_f32<EMBED, 512, 4, 0, 0>
        <<<(NOBJ/16)*(512/64)/4, 128, 0, stream>>>(obj_emb, w_off, b_off, nullptr, offs);

    // 2) deformable sampling + attention
    sampler_kernel<<<NOBJ*NHEAD/8, 256, 0, stream>>>(
        query, key_feat, value, x2d, maskf, obj_xy, strides, img_ind, offs,
        att, v_samples, a_samples, mask_samples, x2d_samples);

    // 3) y1 = att @ w_out + b_out + obj_emb         (2048x256, K=256)
    gemm_wmma_f32<EMBED, EMBED, 4, 0, 1>
        <<<(NOBJ/16)*(EMBED/64)/4, 128, 0, stream>>>(att, w_out, b_out, obj_emb, y1);

    // 4) x1 = LN1(y1)
    layernorm_kernel<<<NOBJ/8, 256, 0, stream>>>(y1, ln1_g, ln1_b, x1);

    // 5) h = relu(x1 @ w1 + b1)                     (2048x1024, K=256)
    gemm_wmma_f32<EMBED, FFND, 4, 1, 0>
        <<<(NOBJ/16)*(FFND/64)/4, 128, 0, stream>>>(x1, w1, b1, nullptr, hbuf);

    // 6) y2 = h @ w2 + b2 + x1                      (2048x256, K=1024)
    gemm_wmma_f32<FFND, EMBED, 4, 0, 1>
        <<<(NOBJ/16)*(EMBED/64)/4, 128, 0, stream>>>(hbuf, w2, b2, x1, y2);

    // 7) out = LN2(y2)
    layernorm_kernel<<<NOBJ/8, 256, 0, stream>>>(y2, ln2_g, ln2_b, out_final);
}